// FeatureAttentionLayer_24318104830052
// MI455X (gfx1250) — compile-verified
//
#include <hip/hip_runtime.h>
#include <cmath>

#define BDIM 8
#define WINDIM 128
#define KDIM 256
#define EDIM 256
#define ALPHA 0.2f
#define PAD 260   // floats; 260 mod 64 = 4 -> 16-lane column reads conflict-free,
                  // 260*4 bytes = 1040 = 16B-aligned rows -> ds_load_b128 legal
#define OPAD 20   // output staging pad (16B-aligned rows, spread banks)

typedef __attribute__((ext_vector_type(2))) float v2f;
typedef __attribute__((ext_vector_type(8))) float v8f;

// -------------------------------------------------------------------------
// Kernel 1: hi = v @ W1^T, hjb = v @ W2^T + b_lin   (v[r][w] = x[b][w][kvar],
// r = b*256 + kvar).  One wave per 16x16 output tile; dual accumulator chains
// of v_wmma_f32_16x16x4_f32 over WIN=128 (32 iterations, A shared).
// grid = 2048 = 128 M-tiles * 16 E-tiles, block = 32 (one wave).
// -------------------------------------------------------------------------
__global__ __launch_bounds__(32)
void k1_hi_hj(const float* __restrict__ x, const float* __restrict__ Wl,
              const float* __restrict__ bl, float* __restrict__ hi_ws,
              float* __restrict__ hjb_ws) {
  const int L     = threadIdx.x;
  const int mtile = blockIdx.x & 127;   // 2048 rows / 16
  const int ntile = blockIdx.x >> 7;    // 256 cols / 16
  const int half  = L >> 4;             // 0 or 1
  const int lan   = L & 15;

  const int rowA = mtile * 16 + lan;    // (b,kvar) flattened
  const int bidx = rowA >> 8;
  const int kvar = rowA & 255;
  const int ecol = ntile * 16 + lan;    // output E column for B/D lanes

  const float* xrow = x + (size_t)bidx * WINDIM * KDIM + kvar; // stride KDIM over w
  const float* wrow = Wl + (size_t)ecol * (2 * WINDIM);        // W_lin row (e)

  v8f chi = {};
  v8f chj = {};
  for (int kk = 0; kk < WINDIM; kk += 4) {
    const int k0 = kk + half * 2;
    v2f a, b1, b2;
    // A (16x4 of v): lane holds row m=lan, K = k0, k0+1
    a.x = xrow[(size_t)k0 * KDIM];
    a.y = xrow[(size_t)(k0 + 1) * KDIM];
    // B (4x16 of W^T): B[k][n] = W[e=n][w=k]
    b1.x = wrow[k0];
    b1.y = wrow[k0 + 1];
    b2.x = wrow[WINDIM + k0];
    b2.y = wrow[WINDIM + k0 + 1];
    chi = __builtin_amdgcn_wmma_f32_16x16x4_f32(false, a, false, b1, (short)0, chi, false, false);
    chj = __builtin_amdgcn_wmma_f32_16x16x4_f32(false, a, false, b2, (short)0, chj, false, false);
  }

  const float be = bl[ecol];
  for (int r = 0; r < 8; ++r) {
    const int M  = r + 8 * half;              // D row
    const int rr = mtile * 16 + M;
    hi_ws [(size_t)rr * EDIM + ecol] = chi[r];
    hjb_ws[(size_t)rr * EDIM + ecol] = chj[r] + be;   // fold b_lin into hj
  }
}

// -------------------------------------------------------------------------
// Kernel 2: fused  e -> softmax -> (attn @ v) -> sigmoid -> transposed store.
// One workgroup per (batch, 16-row i-tile): never materializes the
// (B,K,K,E) 537MB intermediate of the reference.
// grid = B*16 = 128, block = 256 (8 waves).
// -------------------------------------------------------------------------
__global__ __launch_bounds__(256)
void k2_attn(const float* __restrict__ x, const float* __restrict__ hi_ws,
             const float* __restrict__ hjb_ws, const float* __restrict__ a_vec,
             const float* __restrict__ bias_mat, float* __restrict__ out) {
  __shared__ alignas(16) float hi_s[16 * PAD];
  __shared__ alignas(16) float hj_s[16 * PAD];
  __shared__ alignas(16) float e_s [16 * PAD];
  __shared__ alignas(16) float a_s [EDIM];
  __shared__ alignas(16) float an_s[EDIM];     // ALPHA * a
  __shared__ alignas(16) float o_s [WINDIM * OPAD];
  __shared__ float red_s[16 * 16];
  __shared__ float row_s[16];

  const int t  = threadIdx.x;
  const int it = blockIdx.x & 15;   // i tile within K
  const int b  = blockIdx.x >> 4;
  const int i  = t & 15;            // local i row
  const int g  = t >> 4;            // group 0..15 (j lane / staging row)

  // Stage hi tile (16 x 256 = 1024 float4, 4 per thread) and a / alpha*a
  {
    const float4* src4 = (const float4*)(hi_ws + (size_t)(b * KDIM + it * 16) * EDIM);
    for (int c = 0; c < 4; ++c) {
      const int q   = t + 256 * c;  // 0..1023
      const int row = q >> 6;       // 64 float4 per row
      const int c4  = q & 63;
      *(float4*)&hi_s[row * PAD + c4 * 4] = src4[row * 64 + c4];
    }
    const float av = a_vec[t];
    a_s[t]  = av;
    an_s[t] = ALPHA * av;
  }

  // e[i,j] = sum_e a[e]*LeakyReLU(hi[i,e]+hjb[j,e]) + bias[i,j], 16 j rows/block
  for (int jb = 0; jb < 16; ++jb) {
    {
      const float4* src4 = (const float4*)(hjb_ws + (size_t)(b * KDIM + jb * 16) * EDIM);
      for (int c = 0; c < 4; ++c) {
        const int q   = t + 256 * c;
        const int row = q >> 6;
        const int c4  = q & 63;
        *(float4*)&hj_s[row * PAD + c4 * 4] = src4[row * 64 + c4];
      }
    }
    __syncthreads();   // hj_s (and, on jb==0, hi_s/a_s) visible

    float acc = 0.f;
    const float4* h4 = (const float4*)&hi_s[i * PAD];   // 16 rows per wave (same i x2 lanes)
    const float4* j4 = (const float4*)&hj_s[g * PAD];   // broadcast (same g x16 lanes)
    const float4* a4 = (const float4*)a_s;              // broadcast (all lanes)
    const float4* n4 = (const float4*)an_s;
#pragma unroll 4
    for (int e4 = 0; e4 < EDIM / 4; ++e4) {
      const float4 hv = h4[e4];
      const float4 jv = j4[e4];
      const float4 av = a4[e4];
      const float4 nv = n4[e4];
      float z;
      z = hv.x + jv.x; acc = fmaf(z, (z > 0.f) ? av.x : nv.x, acc);
      z = hv.y + jv.y; acc = fmaf(z, (z > 0.f) ? av.y : nv.y, acc);
      z = hv.z + jv.z; acc = fmaf(z, (z > 0.f) ? av.z : nv.z, acc);
      z = hv.w + jv.w; acc = fmaf(z, (z > 0.f) ? av.w : nv.w, acc);
    }
    const int j = jb * 16 + g;
    e_s[i * PAD + j] = acc + bias_mat[(size_t)(it * 16 + i) * KDIM + j];
    __syncthreads();   // protect hj_s restage of next iteration
  }

  // Row softmax over j (256) per i row
  {
    float m = -3.4e38f;
    for (int c = 0; c < 16; ++c) m = fmaxf(m, e_s[i * PAD + g * 16 + c]);
    red_s[i * 16 + g] = m;
  }
  __syncthreads();
  if (t < 16) {
    float m = red_s[t * 16];
    for (int c = 1; c < 16; ++c) m = fmaxf(m, red_s[t * 16 + c]);
    row_s[t] = m;
  }
  __syncthreads();
  {
    const float rm = row_s[i];
    float s = 0.f;
    for (int c = 0; c < 16; ++c) {
      const int j = g * 16 + c;
      const float vexp = __expf(e_s[i * PAD + j] - rm);
      e_s[i * PAD + j] = vexp;
      s += vexp;
    }
    red_s[i * 16 + g] = s;
  }
  __syncthreads();
  if (t < 16) {
    float s = 0.f;
    for (int c = 0; c < 16; ++c) s += red_s[t * 16 + c];
    row_s[t] = 1.0f / s;
  }
  __syncthreads();
  {
    const float rinv = row_s[i];
    for (int c = 0; c < 16; ++c) e_s[i * PAD + g * 16 + c] *= rinv;
  }
  __syncthreads();

  // Fused attn(16x256) @ v(256x128) with f32 WMMA; 8 waves = 8 N-tiles of 16.
  const int wid  = t >> 5;          // wave id -> N tile over WIN
  const int L    = t & 31;
  const int half = L >> 4;
  const int lan  = L & 15;
  const int wcol = wid * 16 + lan;  // output w column

  const float* xcol = x + (size_t)(b * WINDIM + wcol) * KDIM; // v[.][wcol] = x[b][wcol][.]
  v8f c = {};
  for (int kk = 0; kk < KDIM; kk += 4) {
    const int k0 = kk + half * 2;
    v2f a, bmv;
    a.x   = e_s[lan * PAD + k0];      // A: attn row m=lan, K=k0..k0+1
    a.y   = e_s[lan * PAD + k0 + 1];
    bmv.x = xcol[k0];                 // B: v[k0][wcol]
    bmv.y = xcol[k0 + 1];
    c = __builtin_amdgcn_wmma_f32_16x16x4_f32(false, a, false, bmv, (short)0, c, false, false);
  }

  // sigmoid, then stage each wave's 16x16 tile in LDS (wave-private region,
  // LDS in-order per wave -> no barrier) and store coalesced 32B segments.
  for (int r = 0; r < 8; ++r) {
    const int M = r + 8 * half;
    const float val = c[r];
    o_s[wcol * OPAD + M] = 1.0f / (1.0f + __expf(-val));
  }
  {
    const int wrow = wid * 16 + (L >> 1);   // output w row handled by this lane
    const int seg  = (L & 1) * 8;           // float offset within the 16-wide tile
    const float4 v0 = *(const float4*)&o_s[wrow * OPAD + seg];
    const float4 v1 = *(const float4*)&o_s[wrow * OPAD + seg + 4];
    float* dst = out + (size_t)(b * WINDIM + wrow) * KDIM + it * 16 + seg;
    *(float4*)dst       = v0;
    *(float4*)(dst + 4) = v1;
  }
}

// -------------------------------------------------------------------------
extern "C" void kernel_launch(void* const* d_in, const int* in_sizes, int n_in,
                              void* d_out, int out_size, void* d_ws, size_t ws_size,
                              hipStream_t stream) {
  const float* x   = (const float*)d_in[0];   // (8,128,256)
  const float* Wl  = (const float*)d_in[1];   // (256,256)
  const float* bl  = (const float*)d_in[2];   // (256,)
  const float* av  = (const float*)d_in[3];   // (256,1)
  const float* bia = (const float*)d_in[4];   // (256,256)
  float*       out = (float*)d_out;           // (8,128,256)

  float* hi_ws  = (float*)d_ws;                          // 2048 x 256 f32
  float* hjb_ws = hi_ws + (size_t)BDIM * KDIM * EDIM;    // 2048 x 256 f32 (4 MB total)

  k1_hi_hj<<<dim3(2048), dim3(32), 0, stream>>>(x, Wl, bl, hi_ws, hjb_ws);
  k2_attn <<<dim3(BDIM * 16), dim3(256), 0, stream>>>(x, hi_ws, hjb_ws, av, bia, out);
}